// EdgeClassifier_70428873720468
// MI455X (gfx1250) — compile-verified
//
#include <hip/hip_runtime.h>
#include <hip/hip_bf16.h>
#include <math.h>

// GATv2 edge classifier for MI455X (gfx1250, wave32).
// Dense GEMMs on V_WMMA_F32_16X16X4_F32; shared tiles staged to LDS via
// CDNA5 async global->LDS copies (GLOBAL_LOAD_ASYNC_TO_LDS_B128 + ASYNCcnt).

typedef __attribute__((ext_vector_type(2))) float v2f;
typedef __attribute__((ext_vector_type(8))) float v8f;

#define NEG_SLOPE 0.2f

// ---------------------------------------------------------------------------
// CDNA5 async global->LDS copy (16B per lane) + ASYNCcnt wait.
// lds_off: LDS byte address (generic pointer truncated to 32 bits -- the ISA
// flat->LDS mapping discards the upper bits). g: 16B-aligned global pointer.
// ---------------------------------------------------------------------------
__device__ __forceinline__ void async_ld_b128(unsigned lds_off, const float* g)
{
    asm volatile("global_load_async_to_lds_b128 %0, %1, off"
                 :: "v"(lds_off), "v"(g) : "memory");
}

__device__ __forceinline__ void wait_async_all()
{
#if __has_builtin(__builtin_amdgcn_s_wait_asynccnt)
    __builtin_amdgcn_s_wait_asynccnt(0);
#else
    asm volatile("s_wait_asynccnt 0x0" ::: "memory");
#endif
}

__device__ __forceinline__ unsigned lds_addr32(const void* p)
{
    return (unsigned)(unsigned long long)p;
}

// ---------------------------------------------------------------------------
// One wave computes a 16x16 fp32 tile of A[16,K] @ B[K,16] with WMMA f32 K=4.
// A layout (16x4): lanes 0-15 -> M=lane, K={k0,k0+1}; lanes 16-31 -> K={k0+2,k0+3}
// B layout (4x16): lanes 0-15 -> N=lane, rows {k0,k0+1}; lanes 16-31 -> rows {k0+2,k0+3}
// D layout: VGPR r: lanes 0-15 -> M=r, N=lane; lanes 16-31 -> M=8+r, N=lane-16
// ---------------------------------------------------------------------------
template <int K>
__device__ __forceinline__ v8f wmma_f32_tile(const float* __restrict__ A, int lda,
                                             const float* __restrict__ B, int ldb)
{
    const int lane = threadIdx.x & 31;
    const int half = lane >> 4;
    const int l16  = lane & 15;
    v8f c = {};
#pragma unroll
    for (int k0 = 0; k0 < K; k0 += 4) {
        const int ka = k0 + 2 * half;
        v2f a, b;
        a.x = A[l16 * lda + ka];
        a.y = A[l16 * lda + ka + 1];
        b.x = B[ka * ldb + l16];
        b.y = B[(ka + 1) * ldb + l16];
        c = __builtin_amdgcn_wmma_f32_16x16x4_f32(false, a, false, b,
                                                  (short)0, c, false, false);
    }
    return c;
}

// ---------------------------------------------------------------------------
// Y[M,NCOLS] = X[M,K] @ W[K,NCOLS] + bias (+ optional ReLU).
// One block = 16 rows; wave w handles cols [16w,16w+16). A tile staged to LDS
// once per block via async copy (kills the NW-way redundant A traffic).
// ---------------------------------------------------------------------------
template <int K, int NCOLS, bool RELU>
__global__ void __launch_bounds__(32 * (NCOLS / 16))
gemm_bias_kernel(const float* __restrict__ X, const float* __restrict__ W,
                 const float* __restrict__ bias, float* __restrict__ Y, int M)
{
    __shared__ __align__(16) float s_A[16 * K];

    const int rowBase = blockIdx.x * 16;
    const int nthr    = 32 * (NCOLS / 16);

    // Stage A[16,K] -> LDS, 16B per lane per async op. Row clamp keeps loads
    // in-bounds without predication (M=50000 is a multiple of 16 anyway).
    for (int idx4 = threadIdx.x; idx4 < (16 * K) / 4; idx4 += nthr) {
        const int row = idx4 / (K / 4);
        const int col = (idx4 - row * (K / 4)) * 4;
        int gRow = rowBase + row;
        if (gRow >= M) gRow = M - 1;
        async_ld_b128(lds_addr32(&s_A[row * K + col]),
                      X + (size_t)gRow * K + col);
    }
    wait_async_all();
    __syncthreads();

    const int wave    = threadIdx.x >> 5;
    const int colBase = wave * 16;
    const int lane    = threadIdx.x & 31;
    const int half    = lane >> 4;
    const int l16     = lane & 15;

    v8f c = {};
#pragma unroll
    for (int k0 = 0; k0 < K; k0 += 4) {
        const int ka = k0 + 2 * half;
        v2f a, b;
        a.x = s_A[l16 * K + ka];
        a.y = s_A[l16 * K + ka + 1];
        b.x = W[(size_t)ka * NCOLS + colBase + l16];
        b.y = W[(size_t)(ka + 1) * NCOLS + colBase + l16];
        c = __builtin_amdgcn_wmma_f32_16x16x4_f32(false, a, false, b,
                                                  (short)0, c, false, false);
    }

    const int col = colBase + l16;
    const float bv = bias[col];
#pragma unroll
    for (int r = 0; r < 8; ++r) {
        const int row = rowBase + r + 8 * half;
        if (row < M) {
            float v = c[r] + bv;
            if (RELU) v = fmaxf(v, 0.f);
            Y[(size_t)row * NCOLS + col] = v;
        }
    }
}

// ---------------------------------------------------------------------------
// Encoder layer 1: h = relu(x[N,3] @ W1[3,64] + b1)  (K=3, plain VALU)
// ---------------------------------------------------------------------------
__global__ void enc1_kernel(const float* __restrict__ x, const float* __restrict__ W1,
                            const float* __restrict__ b1, float* __restrict__ h, int Nn)
{
    const int t = blockIdx.x * blockDim.x + threadIdx.x;
    if (t >= Nn * 64) return;
    const int i = t >> 6, c = t & 63;
    float s = b1[c];
#pragma unroll
    for (int k = 0; k < 3; ++k) s += x[i * 3 + k] * W1[k * 64 + c];
    h[t] = fmaxf(s, 0.f);
}

// ---------------------------------------------------------------------------
// Per-conv scratch init: emax=-inf, den=0, agg=0
// ---------------------------------------------------------------------------
__global__ void conv_init_kernel(float* __restrict__ emax, float* __restrict__ den,
                                 float* __restrict__ agg, int Nn)
{
    const int t = blockIdx.x * blockDim.x + threadIdx.x;
    if (t < Nn * 2) { emax[t] = -__builtin_inff(); den[t] = 0.f; }
    if (t < Nn * 128) agg[t] = 0.f;
}

__device__ __forceinline__ void atomicMaxF32(float* addr, float v)
{
    if (v >= 0.f) atomicMax((int*)addr, __float_as_int(v));
    else          atomicMin((unsigned int*)addr, __float_as_uint(v));
}

// ---------------------------------------------------------------------------
// e[e,h] = sum_c leaky_relu(xl[src,h,c] + xr[dst,h,c]) * att[h,c]; segment max
// Edges e >= E are self loops (src=dst=e-E).
// ---------------------------------------------------------------------------
__global__ void edge_scores_kernel(const float* __restrict__ xl, const float* __restrict__ xr,
                                   const float* __restrict__ att, const int* __restrict__ eIdx,
                                   float* __restrict__ escore, float* __restrict__ emax,
                                   int E, int Etot)
{
    const int t = blockIdx.x * blockDim.x + threadIdx.x;
    if (t >= Etot * 2) return;
    const int e = t >> 1, h = t & 1;
    int s, d;
    if (e < E) { s = eIdx[e]; d = eIdx[E + e]; } else { s = d = e - E; }
    const float* pl = xl + (size_t)s * 128 + h * 64;
    const float* pr = xr + (size_t)d * 128 + h * 64;
    const float* pa = att + h * 64;
    float acc = 0.f;
#pragma unroll 8
    for (int c = 0; c < 64; ++c) {
        float m = pl[c] + pr[c];
        m = (m > 0.f) ? m : NEG_SLOPE * m;
        acc += m * pa[c];
    }
    escore[t] = acc;
    atomicMaxF32(&emax[d * 2 + h], acc);
}

// ex = exp(e - emax[dst]); den[dst] += ex
__global__ void edge_exp_kernel(const int* __restrict__ eIdx, float* __restrict__ escore,
                                const float* __restrict__ emax, float* __restrict__ den,
                                int E, int Etot)
{
    const int t = blockIdx.x * blockDim.x + threadIdx.x;
    if (t >= Etot * 2) return;
    const int e = t >> 1, h = t & 1;
    const int d = (e < E) ? eIdx[E + e] : (e - E);
    const float ex = __expf(escore[t] - emax[d * 2 + h]);
    escore[t] = ex;
    atomicAdd(&den[d * 2 + h], ex);
}

// agg[dst,h,c] += (ex / (den[dst,h] + 1e-16)) * xl[src,h,c]
__global__ void edge_agg_kernel(const float* __restrict__ xl, const int* __restrict__ eIdx,
                                const float* __restrict__ escore, const float* __restrict__ den,
                                float* __restrict__ agg, int E, int Etot)
{
    const int t = blockIdx.x * blockDim.x + threadIdx.x;   // Etot*128 < 2^31
    if (t >= Etot * 128) return;
    const int e = t >> 7, j = t & 127, h = j >> 6;
    int s, d;
    if (e < E) { s = eIdx[e]; d = eIdx[E + e]; } else { s = d = e - E; }
    const float alpha = escore[e * 2 + h] / (den[d * 2 + h] + 1e-16f);
    atomicAdd(&agg[(size_t)d * 128 + j], alpha * xl[(size_t)s * 128 + j]);
}

// xa = relu(mean over heads + bias); xs += xa
__global__ void head_mean_kernel(const float* __restrict__ agg, const float* __restrict__ bias,
                                 float* __restrict__ xa, float* __restrict__ xs, int Nn)
{
    const int t = blockIdx.x * blockDim.x + threadIdx.x;
    if (t >= Nn * 64) return;
    const int i = t >> 6, c = t & 63;
    float v = 0.5f * (agg[(size_t)i * 128 + c] + agg[(size_t)i * 128 + 64 + c]) + bias[c];
    v = fmaxf(v, 0.f);
    xa[t] = v;
    xs[t] += v;
}

// ---------------------------------------------------------------------------
// Edge MLP: 16 edges per block, 4 waves. ef[16,132] gathered straight into
// LDS with async b128 copies (row stride 528B and all segment boundaries are
// 16B-aligned), then WMMA 132->64 (relu), 64->32 (relu), scalar 32->1.
// ---------------------------------------------------------------------------
__global__ void __launch_bounds__(128)
edge_mlp_kernel(const float* __restrict__ xs, const int* __restrict__ eIdx,
                const float* __restrict__ eAttr,
                const float* __restrict__ W1, const float* __restrict__ b1,
                const float* __restrict__ W2, const float* __restrict__ b2,
                const float* __restrict__ W3, const float* __restrict__ b3,
                float* __restrict__ out, int E)
{
    __shared__ __align__(16) float s_ef[16 * 132];
    __shared__ float s_h1[16 * 64];
    __shared__ float s_h2[16 * 32];

    const int eBase = blockIdx.x * 16;

    // Async gather: ef = [xs[src] | xs[dst] | eAttr], 16B per lane per op.
    for (int idx4 = threadIdx.x; idx4 < (16 * 132) / 4; idx4 += 128) {
        const int row = idx4 / 33;               // 33 x 4-float groups per row
        const int col = (idx4 - row * 33) * 4;
        const int e   = eBase + row;
        const float* src;
        if (col < 64)       src = xs + (size_t)eIdx[e] * 64 + col;
        else if (col < 128) src = xs + (size_t)eIdx[E + e] * 64 + (col - 64);
        else                src = eAttr + (size_t)e * 4;
        async_ld_b128(lds_addr32(&s_ef[row * 132 + col]), src);
    }
    wait_async_all();
    __syncthreads();

    const int wave = threadIdx.x >> 5;
    const int lane = threadIdx.x & 31;
    const int half = lane >> 4, l16 = lane & 15;

    {   // h1 = relu(ef @ W1[132,64] + b1): each wave does 16 columns
        v8f c = wmma_f32_tile<132>(s_ef, 132, W1 + wave * 16, 64);
        const int col = wave * 16 + l16;
        const float bv = b1[col];
#pragma unroll
        for (int r = 0; r < 8; ++r)
            s_h1[(r + 8 * half) * 64 + col] = fmaxf(c[r] + bv, 0.f);
    }
    __syncthreads();

    if (wave < 2) {   // h2 = relu(h1 @ W2[64,32] + b2): waves 0,1
        v8f c = wmma_f32_tile<64>(s_h1, 64, W2 + wave * 16, 32);
        const int col = wave * 16 + l16;
        const float bv = b2[col];
#pragma unroll
        for (int r = 0; r < 8; ++r)
            s_h2[(r + 8 * half) * 32 + col] = fmaxf(c[r] + bv, 0.f);
    }
    __syncthreads();

    if (threadIdx.x < 16) {   // out = h2 @ W3[32,1] + b3
        const int row = threadIdx.x;
        float s = b3[0];
#pragma unroll
        for (int k = 0; k < 32; ++k) s += s_h2[row * 32 + k] * W3[k];
        out[eBase + row] = s;
    }
}

// ---------------------------------------------------------------------------
// Host-side orchestration. Param order = JAX pytree flatten (dict keys sorted):
//  0:x  1:eIndex  2:eAttributes
//  3..8   conv1 {Wl,Wr,att,bias,bl,br}
//  9..14  conv2   15..20 conv3
//  21..26 emlp {W1,W2,W3,b1,b2,b3}
//  27..30 encoder {W1,W2,b1,b2}
// ---------------------------------------------------------------------------
extern "C" void kernel_launch(void* const* d_in, const int* in_sizes, int n_in,
                              void* d_out, int out_size, void* d_ws, size_t ws_size,
                              hipStream_t stream)
{
    const float* x     = (const float*)d_in[0];
    const int*   eIdx  = (const int*)d_in[1];
    const float* eAttr = (const float*)d_in[2];
    const float* convP[3][6];
    for (int l = 0; l < 3; ++l)
        for (int j = 0; j < 6; ++j)
            convP[l][j] = (const float*)d_in[3 + l * 6 + j];
    const float* mW1 = (const float*)d_in[21];
    const float* mW2 = (const float*)d_in[22];
    const float* mW3 = (const float*)d_in[23];
    const float* mb1 = (const float*)d_in[24];
    const float* mb2 = (const float*)d_in[25];
    const float* mb3 = (const float*)d_in[26];
    const float* encW1 = (const float*)d_in[27];
    const float* encW2 = (const float*)d_in[28];
    const float* encb1 = (const float*)d_in[29];
    const float* encb2 = (const float*)d_in[30];

    const int Nn   = in_sizes[0] / 3;
    const int E    = in_sizes[1] / 2;
    const int Etot = E + Nn;

    float* ws = (float*)d_ws;
    size_t off = 0;
    float* xs     = ws + off; off += (size_t)Nn * 64;
    float* xa     = ws + off; off += (size_t)Nn * 64;
    float* xl     = ws + off; off += (size_t)Nn * 128;
    float* xr     = ws + off; off += (size_t)Nn * 128;
    float* agg    = ws + off; off += (size_t)Nn * 128;
    float* escore = ws + off; off += (size_t)Etot * 2;
    float* emax   = ws + off; off += (size_t)Nn * 2;
    float* den    = ws + off; off += (size_t)Nn * 2;
    float* henc   = xl;   // reuse xl as encoder hidden (free until conv1)

    const int rowTiles = (Nn + 15) / 16;

    // Encoder
    enc1_kernel<<<(Nn * 64 + 255) / 256, 256, 0, stream>>>(x, encW1, encb1, henc, Nn);
    gemm_bias_kernel<64, 64, false><<<rowTiles, 128, 0, stream>>>(henc, encW2, encb2, xa, Nn);
    hipMemcpyAsync(xs, xa, (size_t)Nn * 64 * sizeof(float),
                   hipMemcpyDeviceToDevice, stream);

    // Three GATv2 layers
    for (int l = 0; l < 3; ++l) {
        const float* Wl  = convP[l][0];
        const float* Wr  = convP[l][1];
        const float* att = convP[l][2];
        const float* bia = convP[l][3];
        const float* bl  = convP[l][4];
        const float* br  = convP[l][5];

        gemm_bias_kernel<64, 128, false><<<rowTiles, 256, 0, stream>>>(xa, Wl, bl, xl, Nn);
        gemm_bias_kernel<64, 128, false><<<rowTiles, 256, 0, stream>>>(xa, Wr, br, xr, Nn);
        conv_init_kernel<<<(Nn * 128 + 255) / 256, 256, 0, stream>>>(emax, den, agg, Nn);
        edge_scores_kernel<<<(Etot * 2 + 255) / 256, 256, 0, stream>>>(
            xl, xr, att, eIdx, escore, emax, E, Etot);
        edge_exp_kernel<<<(Etot * 2 + 255) / 256, 256, 0, stream>>>(
            eIdx, escore, emax, den, E, Etot);
        edge_agg_kernel<<<(int)(((long long)Etot * 128 + 255) / 256), 256, 0, stream>>>(
            xl, eIdx, escore, den, agg, E, Etot);
        head_mean_kernel<<<(Nn * 64 + 255) / 256, 256, 0, stream>>>(agg, bia, xa, xs, Nn);
    }

    // Edge MLP -> logits [E,1]
    edge_mlp_kernel<<<(E + 15) / 16, 128, 0, stream>>>(
        xs, eIdx, eAttr, mW1, mb1, mW2, mb2, mW3, mb3, (float*)d_out, E);
}